// Net_11312943858272
// MI455X (gfx1250) — compile-verified
//
#include <hip/hip_runtime.h>
#include <hip/hip_bf16.h>

typedef float v2f __attribute__((ext_vector_type(2)));
typedef float v8f __attribute__((ext_vector_type(8)));

static constexpr int N_NODES = 50000;
static constexpr int N_EDGES = 800000;
static constexpr int D_FEAT  = 128;
static constexpr int D_HID   = 500;     // valid cols; H padded to LDH
static constexpr int LDH     = 512;     // padded row stride for H (WMMA-friendly)

// ---------------- degree / norm ----------------
__global__ void k_deg_init(float* __restrict__ deg) {
  int n = blockIdx.x * blockDim.x + threadIdx.x;
  if (n < N_NODES) deg[n] = 1.0f;   // self-loop contributes 1
}

__global__ void k_deg_accum(const long long* __restrict__ dst, float* __restrict__ deg) {
  int e = blockIdx.x * blockDim.x + threadIdx.x;
  if (e < N_EDGES)
    __hip_atomic_fetch_add(&deg[(int)dst[e]], 1.0f, __ATOMIC_RELAXED, __HIP_MEMORY_SCOPE_AGENT);
}

__global__ void k_dinv(float* __restrict__ deg) {
  int n = blockIdx.x * blockDim.x + threadIdx.x;
  if (n < N_NODES) deg[n] = rsqrtf(deg[n]);  // deg >= 1 always
}

// ---------------- pack W1 [128,500] -> Wp [64][512] of float2 (K-pairs, zero-padded) ----------------
// Wp[k2*512 + col] = ( W1[2*k2][col], W1[2*k2+1][col] ), 0 for col >= 500.
__global__ void k_packW(const float* __restrict__ W1, float* __restrict__ Wp) {
  int idx = blockIdx.x * blockDim.x + threadIdx.x;   // 64*512 = 32768
  if (idx >= 64 * 512) return;
  int k2  = idx >> 9;
  int col = idx & 511;
  v2f v = {0.0f, 0.0f};
  if (col < D_HID) {
    v.x = W1[(2 * k2)     * D_HID + col];
    v.y = W1[(2 * k2 + 1) * D_HID + col];
  }
  *((v2f*)Wp + idx) = v;
}

// ---------------- GEMM1: H = X @ W1 via V_WMMA_F32_16X16X4_F32 ----------------
// Block = 8 waves = one 16x512 M-row; each wave owns 4 N-tiles (16x64).
// All loads unconditional: A from X (b64), B from packed Wp (b64). Stores unconditional into padded H.
__global__ __launch_bounds__(256) void k_gemm1(const float* __restrict__ X,
                                               const float* __restrict__ Wp,
                                               float* __restrict__ H) {
  const int lane = threadIdx.x & 31;
  const int wave = threadIdx.x >> 5;     // 0..7
  const int m0   = blockIdx.x * 16;      // 3125 blocks, exact
  const int n0   = wave * 64;            // 4 tiles of 16
  const int l16  = lane & 15;
  const int half = lane >> 4;

  const float* arow = X + (m0 + l16) * D_FEAT;
  const v2f*   bp   = (const v2f*)Wp;    // [64][512] K-pair-packed B

  v8f c0 = {}, c1 = {}, c2 = {}, c3 = {};
#pragma unroll 4
  for (int k0 = 0; k0 < D_FEAT; k0 += 4) {
    const int k2 = (k0 >> 1) + half;                 // packed K-pair row for this lane
    v2f a  = *(const v2f*)(arow + k0 + 2 * half);    // A: K = k0+2h, k0+2h+1
    const v2f* br = bp + (k2 << 9) + n0 + l16;
    v2f b0 = br[0];
    v2f b1 = br[16];
    v2f b2 = br[32];
    v2f b3 = br[48];
    c0 = __builtin_amdgcn_wmma_f32_16x16x4_f32(false, a, false, b0, (short)0, c0, false, false);
    c1 = __builtin_amdgcn_wmma_f32_16x16x4_f32(false, a, false, b1, (short)0, c1, false, false);
    c2 = __builtin_amdgcn_wmma_f32_16x16x4_f32(false, a, false, b2, (short)0, c2, false, false);
    c3 = __builtin_amdgcn_wmma_f32_16x16x4_f32(false, a, false, b3, (short)0, c3, false, false);
  }

#pragma unroll
  for (int r = 0; r < 8; ++r) {
    float* hp = H + (m0 + r + 8 * half) * LDH + n0 + l16;  // C/D: VGPR r -> row r+8h, col l16
    hp[0]  = c0[r];
    hp[16] = c1[r];
    hp[32] = c2[r];
    hp[48] = c3[r];
  }
}

// ---------------- out1 init with self-loop term (H stride LDH -> O stride D_HID) ----------------
__global__ void k_out1_init(const float* __restrict__ H, const float* __restrict__ dinv,
                            float* __restrict__ O) {
  int idx = blockIdx.x * blockDim.x + threadIdx.x;     // over 50000*125 float4s
  if (idx >= N_NODES * 125) return;
  int n = idx / 125, c4 = idx - n * 125;
  float s = dinv[n]; s = s * s;
  float4 hv = *(const float4*)(H + n * LDH + c4 * 4);
  float4 o;
  o.x = hv.x * s; o.y = hv.y * s; o.z = hv.z * s; o.w = hv.w * s;
  *(float4*)(O + n * D_HID + c4 * 4) = o;
}

// ---------------- propagation 1: wave per edge, f32 atomics into L2-resident out1 ----------------
__global__ __launch_bounds__(256) void k_prop1(const long long* __restrict__ src,
                                               const long long* __restrict__ dst,
                                               const float* __restrict__ H,
                                               const float* __restrict__ dinv,
                                               float* __restrict__ O) {
  int e = blockIdx.x * 8 + (threadIdx.x >> 5);
  if (e >= N_EDGES) return;
  const int lane = threadIdx.x & 31;
  const int s = (int)src[e];
  const int d = (int)dst[e];
  const float w = dinv[s] * dinv[d];
  const float* hs = H + s * LDH;
  float* od = O + d * D_HID;
  for (int c4 = lane; c4 < 125; c4 += 32) {
    float4 hv = *(const float4*)(hs + c4 * 4);    // global_load_b128 gather
    float* p = od + c4 * 4;
    __hip_atomic_fetch_add(p + 0, hv.x * w, __ATOMIC_RELAXED, __HIP_MEMORY_SCOPE_AGENT);
    __hip_atomic_fetch_add(p + 1, hv.y * w, __ATOMIC_RELAXED, __HIP_MEMORY_SCOPE_AGENT);
    __hip_atomic_fetch_add(p + 2, hv.z * w, __ATOMIC_RELAXED, __HIP_MEMORY_SCOPE_AGENT);
    __hip_atomic_fetch_add(p + 3, hv.w * w, __ATOMIC_RELAXED, __HIP_MEMORY_SCOPE_AGENT);
  }
}

// ---------------- GEMM2 (N=1): fused bias+ReLU+dot, wave-per-node reduction ----------------
__global__ __launch_bounds__(256) void k_gemm2(const float* __restrict__ O1,
                                               const float* __restrict__ b1,
                                               const float* __restrict__ W2,
                                               const float* __restrict__ b2,
                                               const float* __restrict__ dinv,
                                               float* __restrict__ z,
                                               float* __restrict__ out) {
  int n = blockIdx.x * 8 + (threadIdx.x >> 5);
  if (n >= N_NODES) return;
  const int lane = threadIdx.x & 31;
  const float* row = O1 + n * D_HID;
  float acc = 0.0f;
  for (int c4 = lane; c4 < 125; c4 += 32) {
    float4 v  = *(const float4*)(row + c4 * 4);
    float4 bb = *(const float4*)(b1  + c4 * 4);
    float4 w  = *(const float4*)(W2  + c4 * 4);
    float t;
    t = fmaxf(v.x + bb.x, 0.0f); acc = fmaf(t, w.x, acc);
    t = fmaxf(v.y + bb.y, 0.0f); acc = fmaf(t, w.y, acc);
    t = fmaxf(v.z + bb.z, 0.0f); acc = fmaf(t, w.z, acc);
    t = fmaxf(v.w + bb.w, 0.0f); acc = fmaf(t, w.w, acc);
  }
#pragma unroll
  for (int off = 16; off > 0; off >>= 1) acc += __shfl_xor(acc, off, 32);
  if (lane == 0) {
    z[n] = acc;
    float s = dinv[n];
    out[n] = acc * s * s + b2[0];   // self-loop term + bias (also inits d_out)
  }
}

// ---------------- propagation 2: thread per edge, scalar f32 atomics ----------------
__global__ void k_prop2(const long long* __restrict__ src, const long long* __restrict__ dst,
                        const float* __restrict__ z, const float* __restrict__ dinv,
                        float* __restrict__ out) {
  int e = blockIdx.x * blockDim.x + threadIdx.x;
  if (e >= N_EDGES) return;
  const int s = (int)src[e];
  const int d = (int)dst[e];
  __hip_atomic_fetch_add(&out[d], z[s] * dinv[s] * dinv[d],
                         __ATOMIC_RELAXED, __HIP_MEMORY_SCOPE_AGENT);
}

extern "C" void kernel_launch(void* const* d_in, const int* in_sizes, int n_in,
                              void* d_out, int out_size, void* d_ws, size_t ws_size,
                              hipStream_t stream) {
  const float*     x   = (const float*)d_in[0];
  const long long* ei  = (const long long*)d_in[1];   // int64 [2, 800000]
  const float*     W1  = (const float*)d_in[2];
  const float*     b1  = (const float*)d_in[3];
  const float*     W2  = (const float*)d_in[4];
  const float*     b2  = (const float*)d_in[5];
  float*           out = (float*)d_out;

  float* ws  = (float*)d_ws;
  float* h   = ws;                                    // 50000*512 = 25,600,000 f32 (102.4 MB)
  float* o1  = ws + (size_t)25600000;                 // 50000*500 = 25,000,000 f32 (100 MB)
  float* deg = ws + (size_t)50600000;                 // 50,000 f32 (deg -> dinv in place)
  float* z   = deg + N_NODES;                         // 50,000 f32
  float* wp  = z + N_NODES;                           // 64*512*2 = 65,536 f32 (packed W1)

  const long long* src = ei;
  const long long* dst = ei + N_EDGES;

  k_deg_init <<<(N_NODES + 255) / 256, 256, 0, stream>>>(deg);
  k_deg_accum<<<(N_EDGES + 255) / 256, 256, 0, stream>>>(dst, deg);
  k_dinv     <<<(N_NODES + 255) / 256, 256, 0, stream>>>(deg);

  k_packW    <<<(64 * 512 + 255) / 256, 256, 0, stream>>>(W1, wp);
  k_gemm1    <<<N_NODES / 16, 256, 0, stream>>>(x, wp, h);
  k_out1_init<<<(N_NODES * 125 + 255) / 256, 256, 0, stream>>>(h, deg, o1);
  k_prop1    <<<N_EDGES / 8, 256, 0, stream>>>(src, dst, h, deg, o1);

  k_gemm2    <<<(N_NODES + 7) / 8, 256, 0, stream>>>(o1, b1, W2, b2, deg, z, out);
  k_prop2    <<<(N_EDGES + 255) / 256, 256, 0, stream>>>(src, dst, z, deg, out);
}